// MaxPool2d_81106162417838
// MI455X (gfx1250) — compile-verified
//
#include <hip/hip_runtime.h>

// MaxPool2d 2x2 stride 2, NCHW float32.
// Input : (32, 64, 224, 224)  -> 411 MB
// Output: (32, 64, 112, 112)  -> 103 MB
// Streaming, zero-reuse op: ~514 MB @ 23.3 TB/s => ~22 us floor on MI455X.
// Strategy: aligned 128-bit non-temporal loads/stores, one float4 of output
// per thread, divide-free 2D index mapping, full wave32 blocks.

#define IN_H   224
#define IN_W   224
#define OUT_H  112
#define OUT_W  112
#define NC     (32 * 64)
#define QPR    (OUT_W / 4)   // 28 quads (float4s) per output row
#define ROWS_PER_BLOCK 8     // block = 28 x 8 = 224 threads = 7 wave32s

typedef float f4 __attribute__((ext_vector_type(4)));

__global__ __launch_bounds__(QPR * ROWS_PER_BLOCK) void maxpool2x2_nt_kernel(
    const float* __restrict__ in, float* __restrict__ out)
{
    // Divide-free mapping: exact cover, no bounds checks.
    const int qx = threadIdx.x;                                  // 0..27
    const int oh = blockIdx.x * ROWS_PER_BLOCK + threadIdx.y;    // 0..111
    const int nc = blockIdx.y;                                   // 0..2047

    // Input: two consecutive rows, 8 consecutive floats starting at col 8*qx.
    // All offsets are multiples of 16 bytes -> aligned b128 loads.
    const size_t ibase = (size_t)nc * (IN_H * IN_W)
                       + (size_t)(2 * oh) * IN_W
                       + (size_t)(8 * qx);
    const f4* r0 = (const f4*)(in + ibase);           // row 2*oh
    const f4* r1 = (const f4*)(in + ibase + IN_W);    // row 2*oh + 1

    // Non-temporal: data is streamed exactly once; keep it out of L2's way.
    const f4 a0 = __builtin_nontemporal_load(r0 + 0);
    const f4 a1 = __builtin_nontemporal_load(r0 + 1);
    const f4 b0 = __builtin_nontemporal_load(r1 + 0);
    const f4 b1 = __builtin_nontemporal_load(r1 + 1);

    // 2x2 window max: vertical max, then horizontal pair max.
    f4 v0, v1, res;
    v0.x = fmaxf(a0.x, b0.x);  v0.y = fmaxf(a0.y, b0.y);
    v0.z = fmaxf(a0.z, b0.z);  v0.w = fmaxf(a0.w, b0.w);
    v1.x = fmaxf(a1.x, b1.x);  v1.y = fmaxf(a1.y, b1.y);
    v1.z = fmaxf(a1.z, b1.z);  v1.w = fmaxf(a1.w, b1.w);

    res.x = fmaxf(v0.x, v0.y);
    res.y = fmaxf(v0.z, v0.w);
    res.z = fmaxf(v1.x, v1.y);
    res.w = fmaxf(v1.z, v1.w);

    // Aligned b128 NT store of 4 output pixels.
    const size_t obase = (size_t)nc * (OUT_H * OUT_W)
                       + (size_t)oh * OUT_W
                       + (size_t)(4 * qx);
    __builtin_nontemporal_store(res, (f4*)(out + obase));
}

extern "C" void kernel_launch(void* const* d_in, const int* in_sizes, int n_in,
                              void* d_out, int out_size, void* d_ws, size_t ws_size,
                              hipStream_t stream)
{
    (void)in_sizes; (void)n_in; (void)out_size; (void)d_ws; (void)ws_size;
    const float* x = (const float*)d_in[0];
    float* y = (float*)d_out;

    dim3 block(QPR, ROWS_PER_BLOCK, 1);            // 28 x 8 = 224 threads (7 waves)
    dim3 grid(OUT_H / ROWS_PER_BLOCK, NC, 1);      // 14 x 2048 blocks, exact cover
    maxpool2x2_nt_kernel<<<grid, block, 0, stream>>>(x, y);
}